// ROIAlignLayer_30777735643364
// MI455X (gfx1250) — compile-verified
//
#include <hip/hip_runtime.h>
#include <hip/hip_bf16.h>

// ROI crop_and_resize (bilinear, extrapolation 0) for images [8,256,256,256] NHWC,
// boxes [8,300,4] -> out [2400,7,7,256].
//
// Strategy (MI455X / gfx1250, wave32):
//  - memory-bound gather (~0.6 GB traffic -> ~26us floor @ 23.3 TB/s)
//  - separable bilinear: stage 1 gathers each distinct source row once and does the
//    horizontal lerp into LDS (b128 coalesced loads); stage 2 is a small GEMM
//    out[7xN] = Wy[7x14] x H[14xN] done with V_WMMA_F32_16X16X4_F32.

#define CROP   7
#define IMG_H  256
#define IMG_W  256
#define IMG_C  256
#define BOXES_PER_IMG 300
#define NROWS  14            // 2*CROP source-row slots (top/bottom per iy)
#define CCHUNK 128           // channels per pass (2 passes)
#define NCOLH  (CROP*CCHUNK) // 896 H-columns per pass
#define HSTRIDE (NCOLH + 16) // 912; stride%64==16 -> row+2 shifts 32 banks (no conflicts)

typedef float v2f __attribute__((ext_vector_type(2)));
typedef float v8f __attribute__((ext_vector_type(8)));

__global__ __launch_bounds__(256) void ROIAlignLayer_30777735643364_kernel(
    const float* __restrict__ images,
    const float* __restrict__ boxes,
    float* __restrict__ out)
{
    __shared__ float Hld[16 * HSTRIDE];              // 58,368 B
    __shared__ int   sY0[CROP], sY1[CROP], sX0[CROP], sX1[CROP];
    __shared__ float sYl[CROP], sXl[CROP], sVy[CROP], sVx[CROP];

    const int box = blockIdx.x;
    const int img = box / BOXES_PER_IMG;
    const int tid = threadIdx.x;

    // ---- per-box sampling params: threads 0..6 -> y, threads 32..38 -> x ----
    if (tid < CROP) {
        float by1 = boxes[box*4+0], by2 = boxes[box*4+2];
        float hs  = (by2 - by1) * (float)(IMG_H-1) / (float)(CROP-1);
        float iny = by1 * (float)(IMG_H-1) + (float)tid * hs;
        sVy[tid] = (iny >= 0.f && iny <= (float)(IMG_H-1)) ? 1.f : 0.f;
        float fl = floorf(iny);
        sYl[tid] = iny - fl;
        sY0[tid] = (int)fminf(fmaxf(fl,          0.f), (float)(IMG_H-1));
        sY1[tid] = (int)fminf(fmaxf(ceilf(iny),  0.f), (float)(IMG_H-1));
    } else if (tid >= 32 && tid < 32 + CROP) {
        int i = tid - 32;
        float bx1 = boxes[box*4+1], bx2 = boxes[box*4+3];
        float ws  = (bx2 - bx1) * (float)(IMG_W-1) / (float)(CROP-1);
        float inx = bx1 * (float)(IMG_W-1) + (float)i * ws;
        sVx[i] = (inx >= 0.f && inx <= (float)(IMG_W-1)) ? 1.f : 0.f;
        float fl = floorf(inx);
        sXl[i] = inx - fl;
        sX0[i] = (int)fminf(fmaxf(fl,          0.f), (float)(IMG_W-1));
        sX1[i] = (int)fminf(fmaxf(ceilf(inx),  0.f), (float)(IMG_W-1));
    }
    // zero K-padding rows 14,15 once (0 * garbage would make NaN in WMMA)
    for (int i = tid; i < 2*HSTRIDE; i += 256) Hld[14*HSTRIDE + i] = 0.f;
    __syncthreads();

    const int lane = tid & 31;
    const int wv   = tid >> 5;    // 8 wave32s
    const int half = lane >> 4;
    const int m    = lane & 15;

    // ---- A fragments: Wy[16x16] (rows iy 0..6, cols ry 0..13 nonzero) ----
    // 16x4 f32 A layout: lanes0-15 hold M=lane {K=kr,kr+1}; lanes16-31 {K=kr+2,kr+3}
    float afrag[4][2];
    #pragma unroll
    for (int kc = 0; kc < 4; ++kc) {
        #pragma unroll
        for (int j = 0; j < 2; ++j) {
            int ry = kc*4 + 2*half + j;
            float w = 0.f;
            if (m < CROP && ry < NROWS) {
                if (ry == 2*m)          w = (1.f - sYl[m]) * sVy[m];
                else if (ry == 2*m + 1) w = sYl[m] * sVy[m];
            }
            afrag[kc][j] = w;
        }
    }

    const size_t imgBase = (size_t)img * IMG_H * IMG_W * IMG_C;
    float* obase = out + (size_t)box * (CROP * CROP * IMG_C);

    for (int ch0 = 0; ch0 < IMG_C; ch0 += CCHUNK) {
        // ---- stage 1: gather + horizontal lerp into LDS (each source row once) ----
        for (int task = tid; task < NROWS * CROP * (CCHUNK/4); task += 256) {
            int c4 = task & (CCHUNK/4 - 1);
            int t  = task >> 5;               // CCHUNK/4 == 32
            int ix = t % CROP;
            int ry = t / CROP;
            int iy = ry >> 1;
            int yr = (ry & 1) ? sY1[iy] : sY0[iy];
            float xl = sXl[ix], vx = sVx[ix];
            size_t rowb = imgBase + (size_t)yr * IMG_W * IMG_C + ch0 + c4*4;
            float4 a = *(const float4*)(images + rowb + (size_t)sX0[ix] * IMG_C);
            float4 b = *(const float4*)(images + rowb + (size_t)sX1[ix] * IMG_C);
            float4 h;
            h.x = (a.x + (b.x - a.x) * xl) * vx;
            h.y = (a.y + (b.y - a.y) * xl) * vx;
            h.z = (a.z + (b.z - a.z) * xl) * vx;
            h.w = (a.w + (b.w - a.w) * xl) * vx;
            *(float4*)&Hld[ry * HSTRIDE + ix * CCHUNK + c4*4] = h;
        }
        __syncthreads();

        // ---- stage 2: vertical pass as GEMM via V_WMMA_F32_16X16X4_F32 ----
        // D[16x16] += A(Wy 16x4 chunk) x B(H 4x16 chunk), K chunks kc=0..3
        for (int tile = wv; tile < NCOLH/16; tile += 8) {
            int n0 = tile * 16;
            v8f acc = {0.f,0.f,0.f,0.f,0.f,0.f,0.f,0.f};
            #pragma unroll
            for (int kc = 0; kc < 4; ++kc) {
                int kr = kc*4 + 2*half;   // B 4x16: lanes0-15 rows kr,kr+1; lanes16-31 rows kr+2,kr+3
                v2f A = { afrag[kc][0], afrag[kc][1] };
                v2f B = { Hld[kr * HSTRIDE + n0 + m],
                          Hld[(kr + 1) * HSTRIDE + n0 + m] };
                acc = __builtin_amdgcn_wmma_f32_16x16x4_f32(
                        false, A, false, B, (short)0, acc, false, false);
            }
            // D layout: VGPR v, lanes0-15 -> M=v (only M<7 valid), lanes16-31 -> M=v+8 (unused)
            if (half == 0) {
                int ocol = (n0 >> 7) * IMG_C + ch0 + (n0 & (CCHUNK-1)) + m; // ix*256 + c
                #pragma unroll
                for (int v = 0; v < CROP; ++v)
                    obase[(size_t)v * (CROP*IMG_C) + ocol] = acc[v];
            }
        }
        __syncthreads();
    }
}

extern "C" void kernel_launch(void* const* d_in, const int* in_sizes, int n_in,
                              void* d_out, int out_size, void* d_ws, size_t ws_size,
                              hipStream_t stream) {
    const float* images = (const float*)d_in[0];
    const float* boxes  = (const float*)d_in[1];
    float* out = (float*)d_out;
    int nboxes = in_sizes[1] / 4;   // 2400
    ROIAlignLayer_30777735643364_kernel<<<nboxes, 256, 0, stream>>>(images, boxes, out);
}